// Abbott_55018531062011
// MI455X (gfx1250) — compile-verified
//
#include <hip/hip_runtime.h>
#include <stdint.h>

// Abbott STDP step for MI455X (gfx1250).
// Shapes fixed by reference: D=8 delays, B=2 batch, N=2048 neurons.
//   dW_pot[b,e,o] = Xpost[b,o] * (sum_d xbar_pre[d,b,e]*dmap[d,e,o]) * A_p[e,o]
//   dW_dep[b,e,o] = xbar_post[b,o] * (sum_d Xd[d,b,e]*dmap[d,e,o]) * A_d[e,o]
//   W_new = clip(W + dW_pot - dW_dep, 0, WMAX)
// Roofline: ~268 MB traffic vs ~0.35 GFLOP -> bandwidth-bound (~11.5 us floor
// at 23.3 TB/s). Strategy: b128 everywhere, CDNA5 async global->LDS staging of
// the 8 dmap streams (ASYNCcnt), NT hint on dmap (single-use 134 MB stream;
// keeps the ~134 MB of reusable W/A_p/A_d/output lines resident in 192 MB L2
// across graph replays), compile-time shapes so indexing is shift/mask only.

#define ALPHA_P 0.95f
#define ALPHA_D 0.90f
#define WMAX_C  1.0f

constexpr int NDELAY = 8;
constexpr int BATCH  = 2;
constexpr int NNEUR  = 2048;                    // power of two
constexpr int NSHIFT = 11;                      // log2(NNEUR)
constexpr size_t NN  = (size_t)NNEUR * NNEUR;   // 4M
constexpr int CHUNK  = 1024;                    // outputs per block
constexpr int BLKTHREADS = 256;                 // 8 waves (wave32)

__global__ __launch_bounds__(BLKTHREADS) void Abbott_stdp_w_kernel(
    const float* __restrict__ Xd,        // (D,B,N)
    const float* __restrict__ Xpost,     // (B,N)
    const float* __restrict__ xbar_pre,  // (D,B,N)
    const float* __restrict__ xbar_post, // (B,N)
    const float* __restrict__ W,         // (B,N,N)
    const float* __restrict__ dmap,      // (D,N,N)
    const float* __restrict__ Ap,        // (N,N)
    const float* __restrict__ Ad,        // (N,N)
    float* __restrict__ Wprev,           // (B,N,N)
    float* __restrict__ Wnew)            // (B,N,N)
{
    __shared__ float tile[NDELAY * CHUNK];   // 32 KB: tile[d*CHUNK + j]

    // grid = BATCH * NNEUR * (NNEUR/CHUNK); all shifts/masks (constexpr shapes)
    const int blk = blockIdx.x;
    const int c   = blk & 1;                 // half-row index (NNEUR/CHUNK == 2)
    const int row = blk >> 1;                // b*NNEUR + e
    const int e   = row & (NNEUR - 1);
    const int b   = row >> NSHIFT;
    const int t   = threadIdx.x;
    const int o   = c * CHUNK + t * 4;       // this thread's 4 outputs

    // --- CDNA5 async stage: copy dmap[d, e, o..o+3] for all 8 delays into LDS.
    // VDST = LDS byte offset (flat ptr truncated to 32b == LDS offset in the
    // LDS aperture), VADDR = 64-bit global address, saddr = off (GV mode).
    // th:TH_LOAD_NT -> dmap is a single-use 134 MB stream; don't pollute L2.
#pragma unroll
    for (int d = 0; d < NDELAY; ++d) {
        uint64_t gaddr = (uint64_t)(uintptr_t)(dmap + (size_t)d * NN +
                                               (size_t)e * NNEUR + (size_t)o);
        uint32_t laddr = (uint32_t)(uintptr_t)&tile[d * CHUNK + t * 4];
        asm volatile("global_load_async_to_lds_b128 %0, %1, off th:TH_LOAD_NT"
                     :: "v"(laddr), "v"(gaddr) : "memory");
    }

    // Uniform per-block broadcasts (b,e derived from blockIdx -> SMEM loads).
    float xp[NDELAY], xs[NDELAY];
#pragma unroll
    for (int d = 0; d < NDELAY; ++d) {
        const int idx = ((d << 1) + b) * NNEUR + e;   // (d*BATCH + b)*NNEUR + e
        xp[d] = xbar_pre[idx];
        xs[d] = Xd[idx];
    }

    // Per-output b128 loads overlapping the async copies.
    const size_t wOff = (size_t)b * NN + ((size_t)e << NSHIFT) + (size_t)o;
    const float4 w4  = *(const float4*)(W  + wOff);
    const float4 ap4 = *(const float4*)(Ap + ((size_t)e << NSHIFT) + o);
    const float4 ad4 = *(const float4*)(Ad + ((size_t)e << NSHIFT) + o);
    const float4 po4 = *(const float4*)(Xpost     + ((size_t)b << NSHIFT) + o);
    const float4 xo4 = *(const float4*)(xbar_post + ((size_t)b << NSHIFT) + o);

    // Wait for this wave's async copies (each thread consumes only its own
    // LDS slots -> no workgroup barrier required; ASYNCcnt is per-wave).
    asm volatile("s_wait_asynccnt 0" ::: "memory");

    float pp0 = 0.f, pp1 = 0.f, pp2 = 0.f, pp3 = 0.f;   // sum_d xbar_pre*dmap
    float dd0 = 0.f, dd1 = 0.f, dd2 = 0.f, dd3 = 0.f;   // sum_d Xd*dmap
#pragma unroll
    for (int d = 0; d < NDELAY; ++d) {
        const float4 m = *(const float4*)&tile[d * CHUNK + t * 4]; // ds_load_b128
        pp0 = fmaf(xp[d], m.x, pp0);  dd0 = fmaf(xs[d], m.x, dd0);
        pp1 = fmaf(xp[d], m.y, pp1);  dd1 = fmaf(xs[d], m.y, dd1);
        pp2 = fmaf(xp[d], m.z, pp2);  dd2 = fmaf(xs[d], m.z, dd2);
        pp3 = fmaf(xp[d], m.w, pp3);  dd3 = fmaf(xs[d], m.w, dd3);
    }

    float4 wn;
    wn.x = fminf(fmaxf(w4.x + po4.x * pp0 * ap4.x - xo4.x * dd0 * ad4.x, 0.f), WMAX_C);
    wn.y = fminf(fmaxf(w4.y + po4.y * pp1 * ap4.y - xo4.y * dd1 * ad4.y, 0.f), WMAX_C);
    wn.z = fminf(fmaxf(w4.z + po4.z * pp2 * ap4.z - xo4.z * dd2 * ad4.z, 0.f), WMAX_C);
    wn.w = fminf(fmaxf(w4.w + po4.w * pp3 * ap4.w - xo4.w * dd3 * ad4.w, 0.f), WMAX_C);

    *(float4*)(Wprev + wOff) = w4;   // W_prev passthrough
    *(float4*)(Wnew  + wOff) = wn;
}

// Exponential trace filters: xbar <- alpha*xbar + (1-alpha)*x   (36,864 elems)
__global__ __launch_bounds__(BLKTHREADS) void Abbott_stdp_trace_kernel(
    const float* __restrict__ Xd,        // (D,B,N)
    const float* __restrict__ Xpost,     // (B,N)
    const float* __restrict__ xbar_pre,  // (D,B,N)
    const float* __restrict__ xbar_post, // (B,N)
    float* __restrict__ xpre_new,        // (D,B,N)
    float* __restrict__ xpost_new)       // (B,N)
{
    constexpr int nPre  = NDELAY * BATCH * NNEUR;   // 32768
    constexpr int nPost = BATCH * NNEUR;            // 4096
    const int i = blockIdx.x * BLKTHREADS + threadIdx.x;
    if (i < nPre)
        xpre_new[i] = fmaf(ALPHA_P, xbar_pre[i], (1.0f - ALPHA_P) * Xd[i]);
    if (i < nPost)
        xpost_new[i] = fmaf(ALPHA_D, xbar_post[i], (1.0f - ALPHA_D) * Xpost[i]);
}

extern "C" void kernel_launch(void* const* d_in, const int* in_sizes, int n_in,
                              void* d_out, int out_size, void* d_ws, size_t ws_size,
                              hipStream_t stream) {
    (void)in_sizes; (void)n_in; (void)out_size; (void)d_ws; (void)ws_size;

    const float* Xd        = (const float*)d_in[0];
    const float* Xpost     = (const float*)d_in[1];
    const float* xbar_pre  = (const float*)d_in[2];
    const float* xbar_post = (const float*)d_in[3];
    const float* W         = (const float*)d_in[4];
    const float* dmap      = (const float*)d_in[5];
    const float* Ap        = (const float*)d_in[6];
    const float* Ad        = (const float*)d_in[7];

    // d_out layout: [W_prev | W_new | xbar_pre_new | xbar_post_new]
    float* out       = (float*)d_out;
    float* Wprev     = out;
    float* Wnew      = Wprev + (size_t)BATCH * NN;
    float* xpre_new  = Wnew  + (size_t)BATCH * NN;
    float* xpost_new = xpre_new + (size_t)NDELAY * BATCH * NNEUR;

    const int grid = BATCH * NNEUR * (NNEUR / CHUNK);   // 8192 blocks
    Abbott_stdp_w_kernel<<<grid, BLKTHREADS, 0, stream>>>(
        Xd, Xpost, xbar_pre, xbar_post, W, dmap, Ap, Ad, Wprev, Wnew);

    const int nPre = NDELAY * BATCH * NNEUR;            // 32768
    Abbott_stdp_trace_kernel<<<(nPre + BLKTHREADS - 1) / BLKTHREADS,
                               BLKTHREADS, 0, stream>>>(
        Xd, Xpost, xbar_pre, xbar_post, xpre_new, xpost_new);
}